// HyperLearningLayer_74912819576962
// MI455X (gfx1250) — compile-verified
//
#include <hip/hip_runtime.h>
#include <hip/hip_bf16.h>

// ---------------------------------------------------------------------------
// HyperLearningLayer for MI455X (gfx1250, wave32).
//
// Key math: attention has a single kv token -> softmax over a length-1 axis
// is identically 1 -> out = broadcast(v). q/Wq/Wk are dead. Layer reduces to:
//   pooled[mod][b][c] = mean_{24^3}(feats[b][c])                 [2][8][128]
//   v[b]     = pooled_m[b] @ Wv_mri^T + pooled_p[b] @ Wv_pet^T   [8][512]
//   delta[b] = v[b] @ Wout^T + bout                              [8][512]
//   out[b][n][:] = h_latent[b][n][:] + delta[b][:]
//
// Bandwidth-dominated (~147 MB total traffic ~= 6.3 us @ 23.3 TB/s).
// Tiny f32 GEMMs use V_WMMA_F32_16X16X4_F32 (exact fp32).
// Pooled/delta are zero-padded to 16 rows so the WMMA loops are branch-free
// (EXEC stays all-1s, fragment loads are unconditional 8-byte loads).
// Streaming traffic uses non-temporal hints (native clang vectors, since the
// builtins reject HIP_vector_type structs).
// ---------------------------------------------------------------------------

typedef __attribute__((ext_vector_type(2))) float v2f;
typedef __attribute__((ext_vector_type(4))) float v4f;
typedef __attribute__((ext_vector_type(8))) float v8f;

#define VOX 13824          // 24*24*24
#define VOX4 3456          // VOX / 4
#define CIN 128
#define DIM 512
#define BATCH 8
#define MPAD 16            // WMMA M padding

// Workspace layout (floats):
//   pooled : [2][16][128]  (rows 8..15 zeroed)   -> 4096
//   v_buf  : [16][512]                           -> 8192
//   delta  : [16][512]     (rows 8..15 unused)   -> 8192
#define WS_POOLED 0
#define WS_VBUF   4096
#define WS_DELTA  (4096 + 8192)

// --------------------------------------------------------------------------
// Kernel 0: zero the padding rows of pooled ([mod][8..15][128] = 2048 floats).
// --------------------------------------------------------------------------
__global__ void pad_zero_kernel(float* __restrict__ pooled) {
    const int t   = blockIdx.x * blockDim.x + threadIdx.x;   // 0..2047
    const int mod = t >> 10;
    const int i   = t & 1023;
    pooled[mod * (MPAD * CIN) + BATCH * CIN + i] = 0.0f;
}

// --------------------------------------------------------------------------
// Kernel 1: pooled[mod*2048 + b*128 + c] = mean over 13824 voxels.
// One block (256 threads = 8 waves) per (mod,b,c) row: 2048 blocks.
// Non-temporal 16B loads (read-once stream, 113 MB total).
// --------------------------------------------------------------------------
__global__ void pool_kernel(const float* __restrict__ mri,
                            const float* __restrict__ pet,
                            float* __restrict__ pooled) {
    const int id  = blockIdx.x;            // 0..2047
    const int mod = id >> 10;              // 0 = mri, 1 = pet
    const int bc  = id & 1023;             // b*128 + c
    const v4f* src = (const v4f*)((mod ? pet : mri) + (size_t)bc * VOX);

    float s = 0.0f;
    for (int i = threadIdx.x; i < VOX4; i += blockDim.x) {
        v4f t = __builtin_nontemporal_load(&src[i]);
        s += (t.x + t.y) + (t.z + t.w);
    }

    __shared__ float red[256];
    red[threadIdx.x] = s;
    __syncthreads();
    for (int off = 128; off > 0; off >>= 1) {
        if (threadIdx.x < (unsigned)off) red[threadIdx.x] += red[threadIdx.x + off];
        __syncthreads();
    }
    if (threadIdx.x == 0)
        pooled[mod * (MPAD * CIN) + bc] = red[0] * (1.0f / (float)VOX);
}

// --------------------------------------------------------------------------
// Kernel 2: v[16][512] = pooled_m(16x128) @ Wv_mri^T + pooled_p(16x128) @ Wv_pet^T
// One wave (32 threads) per 16-column output tile -> 32 blocks.
// V_WMMA_F32_16X16X4_F32 fragment layouts (ISA 7.12.2):
//   A 16x4 : lanes 0-15 (M=lane, K=k,k+1), lanes 16-31 K=k+2,k+3
//   B 4x16 : lanes 0-15 (N=lane, K=k,k+1), lanes 16-31 K=k+2,k+3
//   C/D    : VGPR r -> M=r (lanes 0-15) / M=r+8 (lanes 16-31), N=lane&15
// --------------------------------------------------------------------------
__global__ void gemmA_kernel(const float* __restrict__ pooled,  // [2][16][128]
                             const float* __restrict__ Wv_m,    // [512][128]
                             const float* __restrict__ Wv_p,    // [512][128]
                             float* __restrict__ v_out) {       // [16][512]
    const int lane   = threadIdx.x;        // 0..31
    const int tile_n = blockIdx.x;         // 0..31
    const int half   = lane >> 4;          // 0: lanes 0-15, 1: lanes 16-31
    const int l15    = lane & 15;
    const int n      = tile_n * 16 + l15;  // output column == W row

    v8f acc = {};
    for (int pass = 0; pass < 2; ++pass) {
        const float* A = pooled + pass * (MPAD * CIN);   // 16x128, rows 8..15 = 0
        const float* W = pass ? Wv_p : Wv_m;
        for (int k = 0; k < CIN; k += 4) {
            const int ka = k + half * 2;                 // even -> 8B aligned
            const v2f a = *(const v2f*)(A + l15 * CIN + ka);
            const v2f b = *(const v2f*)(W + n * CIN + ka);
            acc = __builtin_amdgcn_wmma_f32_16x16x4_f32(
                false, a, false, b, (short)0, acc, false, false);
        }
    }
    #pragma unroll
    for (int r = 0; r < 8; ++r) {
        const int m = r + half * 8;
        v_out[m * DIM + tile_n * 16 + l15] = acc[r];
    }
}

// --------------------------------------------------------------------------
// Kernel 3: delta[16][512] = v(16x512) @ Wout^T + bout (rows 8..15 garbage-ok)
// --------------------------------------------------------------------------
__global__ void gemmB_kernel(const float* __restrict__ v_in,   // [16][512]
                             const float* __restrict__ Wout,   // [512][512]
                             const float* __restrict__ bout,   // [512]
                             float* __restrict__ delta) {      // [16][512]
    const int lane   = threadIdx.x;
    const int tile_n = blockIdx.x;         // 0..31
    const int half   = lane >> 4;
    const int l15    = lane & 15;
    const int n      = tile_n * 16 + l15;

    v8f acc = {};
    for (int k = 0; k < DIM; k += 4) {
        const int ka = k + half * 2;
        const v2f a = *(const v2f*)(v_in + l15 * DIM + ka);
        const v2f b = *(const v2f*)(Wout + n * DIM + ka);
        acc = __builtin_amdgcn_wmma_f32_16x16x4_f32(
            false, a, false, b, (short)0, acc, false, false);
    }
    const float bias = bout[tile_n * 16 + l15];
    #pragma unroll
    for (int r = 0; r < 8; ++r) {
        const int m = r + half * 8;
        delta[m * DIM + tile_n * 16 + l15] = acc[r] + bias;
    }
}

// --------------------------------------------------------------------------
// Kernel 4: out[b][n][d] = h_latent[b][n][d] + delta[b][d]   (16B NT lds/sts)
// 8*1024*512 floats = 1,048,576 vec4 -> 4096 blocks x 256.
// --------------------------------------------------------------------------
__global__ void add_kernel(const float* __restrict__ h,
                           const float* __restrict__ delta,
                           float* __restrict__ out) {
    const size_t i4 = (size_t)blockIdx.x * blockDim.x + threadIdx.x;
    const size_t e  = i4 * 4;
    const int b = (int)(e >> 19);          // / (1024*512)
    const int d = (int)(e & 511);          // % 512 (512 % 4 == 0 so vec4-safe)
    const v4f hv = __builtin_nontemporal_load(&((const v4f*)h)[i4]);
    const v4f dv = *(const v4f*)(delta + b * DIM + d);   // hot, cache it
    const v4f o  = hv + dv;
    __builtin_nontemporal_store(o, &((v4f*)out)[i4]);
}

// --------------------------------------------------------------------------
extern "C" void kernel_launch(void* const* d_in, const int* in_sizes, int n_in,
                              void* d_out, int out_size, void* d_ws, size_t ws_size,
                              hipStream_t stream) {
    const float* h_latent = (const float*)d_in[0];   // [8][1024][512]
    const float* h_mri    = (const float*)d_in[1];   // [8][128][24][24][24]
    const float* h_pet    = (const float*)d_in[2];   // [8][128][24][24][24]
    // d_in[3] (Wq), d_in[4] (Wk_mri), d_in[5] (Wk_pet): dead — softmax over a
    // singleton axis makes attention weights identically 1.
    const float* Wv_mri   = (const float*)d_in[6];   // [512][128]
    const float* Wv_pet   = (const float*)d_in[7];   // [512][128]
    const float* Wout     = (const float*)d_in[8];   // [512][512]
    const float* bout     = (const float*)d_in[9];   // [512]
    float* out = (float*)d_out;

    float* ws     = (float*)d_ws;
    float* pooled = ws + WS_POOLED;   // [2][16][128]
    float* v_buf  = ws + WS_VBUF;     // [16][512]
    float* delta  = ws + WS_DELTA;    // [16][512]

    pad_zero_kernel<<<8,    256, 0, stream>>>(pooled);
    pool_kernel    <<<2048, 256, 0, stream>>>(h_mri, h_pet, pooled);
    gemmA_kernel   <<<32,   32,  0, stream>>>(pooled, Wv_mri, Wv_pet, v_buf);
    gemmB_kernel   <<<32,   32,  0, stream>>>(v_buf, Wout, bout, delta);
    add_kernel     <<<4096, 256, 0, stream>>>(h_latent, delta, out);
}